// Clahe3D_22067541967498
// MI455X (gfx1250) — compile-verified
//
#include <hip/hip_runtime.h>
#include <math.h>

typedef __attribute__((ext_vector_type(2))) float v2f;
typedef __attribute__((ext_vector_type(8))) float v8f;

#define NB   64
#define GG   8
#define DIMN 128
#define VPT  4096

// ---------- quintic B-spline basis (support |x|<3) ----------
__device__ __forceinline__ float bspline5(float x) {
    float t  = fabsf(x);
    float t2 = t * t, t3 = t2 * t, t4 = t2 * t2, t5 = t4 * t;
    float w0 = 11.0f/20.0f - 0.5f*t2 + 0.25f*t4 - t5*(1.0f/12.0f);
    float w1 = 17.0f/40.0f + t*(5.0f/8.0f) - t2*(7.0f/4.0f) + t3*(5.0f/4.0f)
             - t4*(3.0f/8.0f) + t5*(1.0f/24.0f);
    float u  = 3.0f - t;
    float w2 = (u*u)*(u*u)*u*(1.0f/120.0f);
    return t < 1.0f ? w0 : (t < 2.0f ? w1 : (t < 3.0f ? w2 : 0.0f));
}

__device__ __forceinline__ int reflect_dct2(int i, int n) {
    int p = 2 * n;
    int r = i % p; if (r < 0) r += p;
    return r < n ? r : p - 1 - r;
}

// ---------- K0: axis matrix M[128][8] (same for all 3 axes) + min/max init ----------
__global__ void k_axis(float* __restrict__ M, unsigned* __restrict__ mm) {
    int s = threadIdx.x;                     // 0..127
    if (s == 0) { mm[0] = 0x7f7fffffu; mm[1] = 0u; }
    const float lo = -0.5f - 0.25f / 8.0f;   // -0.53125
    const float hi =  7.0f + 0.5f + 0.25f / 8.0f;
    float c = lo + (hi - lo) * (float)s / 127.0f;
    float row[8];
    #pragma unroll
    for (int k = 0; k < 8; k++) row[k] = 0.0f;
    int base = (int)floorf(c) - 2;
    #pragma unroll
    for (int t = 0; t < 6; t++) {
        int tap = base + t;
        float w = bspline5(c - (float)tap);
        row[reflect_dct2(tap, GG)] += w;
    }
    #pragma unroll
    for (int k = 0; k < 8; k++) M[s * 8 + k] = row[k];
}

// ---------- K1: per-tile KDE histogram -> clip -> redistribute -> CDF ----------
// sigma=0.001: only bins within ~0.0132 of the value are nonzero in fp32
// (exp underflows to exactly 0 beyond that) -> evaluate a 4-bin window.
__global__ void k_hist(const float* __restrict__ x, float* __restrict__ cdf) {
    __shared__ float hist[NB];
    __shared__ float sScale, sResid, sRedist;
    int b = blockIdx.x;                       // tile id, 512 tiles
    int ti = b >> 6, tj = (b >> 3) & 7, tk = b & 7;
    int tid = threadIdx.x;
    for (int k = tid; k < NB; k += 256) hist[k] = 0.0f;
    __syncthreads();

    for (int v = tid; v < VPT; v += 256) {
        int ld = v >> 8, lh = (v >> 4) & 15, lw = v & 15;
        int d = ti * 16 + ld, h = tj * 16 + lh, w = tk * 16 + lw;
        float val = x[(d * DIMN + h) * DIMN + w];
        int k0 = (int)floorf(val * 63.0f);
        #pragma unroll
        for (int t = -1; t <= 2; t++) {
            int k = k0 + t;
            if ((unsigned)k < 64u) {
                float z = (val - (float)k * (1.0f / 63.0f)) * 1000.0f;
                float e = __expf(-0.5f * z * z);
                atomicAdd(&hist[k], e);
            }
        }
    }
    __syncthreads();

    if (tid == 0) {
        float S = 0.0f;
        for (int k = 0; k < NB; k++) S += hist[k];
        // histos = hist * vpt / (S + vpt*1e-10)
        sScale = 4096.0f / (S + 4096.0f * 1e-10f);
    }
    __syncthreads();
    if (tid < NB) hist[tid] = fminf(hist[tid] * sScale, 256.0f);  // limit=floor(4*4096/64)
    __syncthreads();
    if (tid == 0) {
        float sc = 0.0f;
        for (int k = 0; k < NB; k++) sc += hist[k];
        float clipped = 4096.0f - sc;
        float r = fmodf(clipped, 64.0f);
        sResid = r;
        sRedist = (clipped - r) * (1.0f / 64.0f);
        float cum = 0.0f;
        int tb = ti * 64 + tj * 8 + tk;       // grid cell (i,j,k)
        for (int k = 0; k < NB; k++) {
            cum += hist[k] + sRedist + ((float)k < sResid ? 1.0f : 0.0f);
            cdf[k * 512 + tb] = cum * (63.0f / 4096.0f);   // cdf[(n*64+i*8+j)*8 + k_g]
        }
    }
}

// ---------- K2: W-axis spline contraction as WMMA GEMM ----------
// T1[r][w] = sum_k cdf[r][k] * M[w][k],  r=(n,i,j) in 4096 rows, w in 128, K=8.
// 16x16 output tiles, two V_WMMA_F32_16X16X4_F32 per tile (K=4 each).
__global__ void k_spline_w(const float* __restrict__ cdf, const float* __restrict__ M,
                           float* __restrict__ T1) {
    int lane = threadIdx.x & 31;
    int wave = blockIdx.x * (blockDim.x >> 5) + (threadIdx.x >> 5);   // 0..2047
    int tileR = wave >> 3;     // 256 row tiles
    int tileW = wave & 7;      // 8 col tiles
    int half = lane >> 4;      // lane 0-15 vs 16-31
    int l16  = lane & 15;

    v8f acc = {};
    #pragma unroll
    for (int k0 = 0; k0 < 8; k0 += 4) {
        int kk = k0 + (half << 1);                 // K in VGPR0: 0/2 (per lane half)
        int m  = tileR * 16 + l16;                 // A row (per ISA A-layout)
        v2f a; a.x = cdf[m * 8 + kk]; a.y = cdf[m * 8 + kk + 1];
        int wc = tileW * 16 + l16;                 // B col, B[k][w] = M[w][k]
        v2f bv; bv.x = M[wc * 8 + kk]; bv.y = M[wc * 8 + kk + 1];
        acc = __builtin_amdgcn_wmma_f32_16x16x4_f32(
                  false, a, false, bv, (short)0, acc, false, false);
    }
    #pragma unroll
    for (int r = 0; r < 8; r++) {                  // D layout: row = r + 8*half
        int m  = tileR * 16 + r + half * 8;
        int wc = tileW * 16 + l16;
        T1[m * 128 + wc] = acc[r];
    }
}

// ---------- K3: fused D/H-axis contraction + 6-tap bin spline ----------
// One block per d-slice. Q[n][j][w] = sum_i Md[d,i]*T1[n,i,j,w] lives in
// 256 KB of LDS (CDNA5 WGP has 320 KB). Then each voxel: 6 bin taps x 8 FMAs.
__global__ void k_interp(const float* __restrict__ x, const float* __restrict__ Mg,
                         const float* __restrict__ T1, float* __restrict__ out,
                         unsigned* __restrict__ mm) {
    extern __shared__ float Q[];                   // 64*8*128 = 65536 floats = 256 KB
    __shared__ float Msh[DIMN * 8];
    __shared__ unsigned smn, smx;
    int d = blockIdx.x;
    int tid = threadIdx.x;
    if (tid == 0) { smn = 0x7f7fffffu; smx = 0u; }
    for (int i = tid; i < DIMN * 8; i += 256) Msh[i] = Mg[i];
    __syncthreads();

    float md[8];
    #pragma unroll
    for (int i = 0; i < 8; i++) md[i] = Msh[d * 8 + i];

    for (int idx = tid; idx < 65536; idx += 256) { // Q[n*1024 + j*128 + w]
        int n = idx >> 10, j = (idx >> 7) & 7, w = idx & 127;
        const float* t1p = T1 + n * 8192 + j * 128 + w;   // + i*1024
        float q = 0.0f;
        #pragma unroll
        for (int i = 0; i < 8; i++) q += md[i] * t1p[i * 1024];
        Q[idx] = q;
    }
    __syncthreads();

    float lmn = 3.4e38f, lmx = 0.0f;
    for (int v = tid; v < DIMN * DIMN; v += 256) {
        int h = v >> 7, w = v & 127;
        float val = x[(d * DIMN + h) * DIMN + w];
        float cb = val * 63.0f;
        int base = (int)floorf(cb) - 2;
        float acc = 0.0f;
        #pragma unroll
        for (int t = 0; t < 6; t++) {
            int tap = base + t;
            float wb = bspline5(cb - (float)tap);
            int n = reflect_dct2(tap, NB);
            const float* qp = Q + (n << 10) + w;
            float s = 0.0f;
            #pragma unroll
            for (int j = 0; j < 8; j++) s += Msh[h * 8 + j] * qp[j * 128];
            acc += wb * s;
        }
        out[(d * DIMN + h) * DIMN + w] = acc;
        lmn = fminf(lmn, acc); lmx = fmaxf(lmx, acc);
    }
    // all values >= 0 -> uint bit pattern preserves float order
    atomicMin(&smn, __float_as_uint(lmn));
    atomicMax(&smx, __float_as_uint(lmx));
    __syncthreads();
    if (tid == 0) { atomicMin(&mm[0], smn); atomicMax(&mm[1], smx); }
}

// ---------- K4: global normalize ----------
__global__ void k_norm(float* __restrict__ out, const unsigned* __restrict__ mm, int n) {
    int i = blockIdx.x * blockDim.x + threadIdx.x;
    if (i >= n) return;
    float mn = __uint_as_float(mm[0]);
    float mx = __uint_as_float(mm[1]);
    float inv = 1.0f / (mx - mn + 1e-10f);
    out[i] = (out[i] - mn) * inv;
}

extern "C" void kernel_launch(void* const* d_in, const int* in_sizes, int n_in,
                              void* d_out, int out_size, void* d_ws, size_t ws_size,
                              hipStream_t stream) {
    (void)in_sizes; (void)n_in; (void)ws_size;
    const float* x = (const float*)d_in[0];
    float* out = (float*)d_out;

    float* wsf = (float*)d_ws;
    float* M   = wsf;                                // 1024 floats
    float* cdf = wsf + 1024;                         // 32768 floats  [64 bins][512 cells]
    float* T1  = wsf + 1024 + 32768;                 // 524288 floats [n][i][j][w]
    unsigned* mm = (unsigned*)(wsf + 1024 + 32768 + 524288);   // min/max bits

    k_axis    <<<1,    128, 0,      stream>>>(M, mm);
    k_hist    <<<512,  256, 0,      stream>>>(x, cdf);
    k_spline_w<<<256,  256, 0,      stream>>>(cdf, M, T1);     // 2048 waves, 1 WMMA-tile each
    k_interp  <<<128,  256, 262144, stream>>>(x, M, T1, out, mm);  // 256 KB dynamic LDS
    k_norm    <<<(out_size + 255) / 256, 256, 0, stream>>>(out, mm, out_size);
}